// VisitTransformer_6038724018977
// MI455X (gfx1250) — compile-verified
//
#include <hip/hip_runtime.h>
#include <stdint.h>
#include <math.h>

#define MAX_VISITS_HW 510   // hard-coded in the reference module (512 - 2)
#define MAX_DAY_F     364.0f

__device__ __forceinline__ int lower_bound_i32(const int* __restrict__ a, int n, int key) {
  int lo = 0, hi = n;
  while (lo < hi) {
    int mid = (lo + hi) >> 1;
    if (a[mid] < key) lo = mid + 1; else hi = mid;
  }
  return lo;
}

// ---------------------------------------------------------------------------
// K1: fill the entire [B, 510, D] output with pad_embedding using the CDNA5
// async LDS->global store path (ASYNCcnt-tracked, no VGPR data movement).
// ---------------------------------------------------------------------------
__global__ void pad_fill_async_kernel(const float* __restrict__ pad,
                                      float* __restrict__ out,
                                      int d4cnt,          // D/4
                                      long n_chunks)      // out_size/4 float4 chunks
{
  __shared__ float4 spad[256];  // supports D up to 1024
  for (int i = threadIdx.x; i < d4cnt; i += blockDim.x)
    spad[i] = ((const float4*)pad)[i];
  __syncthreads();

  unsigned long long base = (unsigned long long)(uintptr_t)out;
  long tid    = (long)blockIdx.x * blockDim.x + threadIdx.x;
  long stride = (long)gridDim.x * blockDim.x;

  for (long g = tid; g < n_chunks; g += stride) {
    int d4 = (int)(g % (long)d4cnt);
    // low 32 bits of a generic pointer into LDS == workgroup-relative LDS byte
    // address (ISA 10.2 aperture mapping)
    unsigned int lds_off = (unsigned int)(uintptr_t)&spad[d4];
    unsigned int voff    = (unsigned int)(g * 16);   // byte offset < 2^31
    // GVS mode: mem_addr = SGPR_U64 + VGPR_I32; VSRC = LDS byte address
    asm volatile("global_store_async_from_lds_b128 %0, %1, %2"
                 :: "v"(voff), "v"(lds_off), "s"(base) : "memory");
  }
  asm volatile("s_wait_asynccnt 0" ::: "memory");
}

// ---------------------------------------------------------------------------
// K2: one visit per 64-lane group (4 visits per 256-thread block).
// Gather + segment-sum the visit's code embeddings (each row = one coalesced
// 1 KB transaction: 64 lanes x float4), fuse the sinusoidal time embedding,
// and store to the scattered (person, slot) row.
// ---------------------------------------------------------------------------
__global__ void visit_embed_kernel(const int*   __restrict__ all_codes,
                                   const int*   __restrict__ code_to_visit,
                                   const int*   __restrict__ visit_person,
                                   const int*   __restrict__ visit_slot,
                                   const float* __restrict__ times,
                                   const float* __restrict__ concept_emb,
                                   const float* __restrict__ timescales,
                                   float*       __restrict__ out,
                                   int total_codes, int total_visits, int D)
{
  const int group = threadIdx.x >> 6;        // 0..3
  const int lane  = threadIdx.x & 63;        // 0..63
  const int v     = blockIdx.x * 4 + group;
  if (v >= total_visits) return;

  // Segment boundaries in the sorted code_to_visit array (uniform per group).
  const int cstart = lower_bound_i32(code_to_visit, total_codes, v);
  const int cend   = lower_bound_i32(code_to_visit, total_codes, v + 1);

  float t = times[v];
  t = fminf(fmaxf(t, 0.0f), MAX_DAY_F);

  const int person = visit_person[v];
  const int slot   = visit_slot[v];

  const int d4cnt = D >> 2;          // 64 for D=256
  const int half4 = d4cnt >> 1;      // float4 chunks in each sin/cos half
  const float4* __restrict__ ts4 = (const float4*)timescales;
  float4* __restrict__ orow =
      (float4*)(out + ((long)person * MAX_VISITS_HW + slot) * (long)D);

  for (int d4 = lane; d4 < d4cnt; d4 += 64) {
    // segment sum of this visit's code embeddings
    float4 acc = make_float4(0.f, 0.f, 0.f, 0.f);
    for (int c = cstart; c < cend; ++c) {
      const int code = all_codes[c];                        // uniform broadcast
      const float4 e = ((const float4*)(concept_emb + (long)code * D))[d4];
      acc.x += e.x; acc.y += e.y; acc.z += e.z; acc.w += e.w;
    }

    // fused sinusoidal time embedding: [sin(t*ts) | cos(t*ts)]
    float4 te;
    if (d4 < half4) {
      const float4 s = ts4[d4];
      te.x = sinf(t * s.x); te.y = sinf(t * s.y);
      te.z = sinf(t * s.z); te.w = sinf(t * s.w);
    } else {
      const float4 s = ts4[d4 - half4];
      te.x = cosf(t * s.x); te.y = cosf(t * s.y);
      te.z = cosf(t * s.z); te.w = cosf(t * s.w);
    }

    orow[d4] = make_float4(acc.x + te.x, acc.y + te.y,
                           acc.z + te.z, acc.w + te.w);
  }
}

extern "C" void kernel_launch(void* const* d_in, const int* in_sizes, int n_in,
                              void* d_out, int out_size, void* d_ws, size_t ws_size,
                              hipStream_t stream) {
  const int*   all_codes     = (const int*)  d_in[0];
  const int*   code_to_visit = (const int*)  d_in[1];
  const int*   visit_person  = (const int*)  d_in[2];
  const int*   visit_slot    = (const int*)  d_in[3];
  const float* times         = (const float*)d_in[4];
  const float* concept_emb   = (const float*)d_in[5];
  const float* pad_embedding = (const float*)d_in[6];
  const float* timescales    = (const float*)d_in[7];
  float*       out           = (float*)d_out;

  const int total_codes  = in_sizes[1];
  const int total_visits = in_sizes[2];
  const int D            = in_sizes[6];       // embedding dim (256)

  // K1: pad-fill the whole output (visit slots are overwritten by K2)
  const long n_chunks = (long)out_size / 4;   // float4 chunks
  int fill_blocks = (int)((n_chunks + 255) / 256);
  if (fill_blocks > 4096) fill_blocks = 4096;
  pad_fill_async_kernel<<<fill_blocks, 256, 0, stream>>>(
      pad_embedding, out, D >> 2, n_chunks);

  // K2: gather + segment-sum + time embedding + scatter
  const int vis_blocks = (total_visits + 3) / 4;
  visit_embed_kernel<<<vis_blocks, 256, 0, stream>>>(
      all_codes, code_to_visit, visit_person, visit_slot, times,
      concept_emb, timescales, out, total_codes, total_visits, D);
}